// GraphAttentionLayer_55490977465039
// MI455X (gfx1250) — compile-verified
//
#include <hip/hip_runtime.h>
#include <math.h>

#define IN_F    256
#define OUT_F   256
#define NH      4
#define HDIM    64
#define NNODES  20000
#define NEDGES  160000
#define LN_EPSF 1e-5f
#define NSLOPE  0.2f

typedef __attribute__((ext_vector_type(2))) float v2f;
typedef __attribute__((ext_vector_type(8))) float v8f;

// ---- monotonic float <-> ordered-uint encoding (for atomicMax on floats) ----
__device__ __forceinline__ unsigned f2ord(float f) {
    unsigned u = __float_as_uint(f);
    return (u & 0x80000000u) ? ~u : (u | 0x80000000u);
}
__device__ __forceinline__ float ord2f(unsigned u) {
    unsigned v = (u & 0x80000000u) ? (u & 0x7fffffffu) : ~u;
    return __uint_as_float(v);
}

// ---------------------------------------------------------------------------
// 0) init: zero agg accumulator, reset per-head max (encoding 0 == -inf side)
// ---------------------------------------------------------------------------
__global__ void gat_init(float* __restrict__ agg, unsigned* __restrict__ maxb) {
    const size_t i = (size_t)blockIdx.x * 256 + threadIdx.x;
    if (i < (size_t)NNODES * OUT_F) agg[i] = 0.0f;
    if (i < NH) maxb[i] = 0u;
}

// ---------------------------------------------------------------------------
// 1) t[h][n][d] = sum_k X[n][k] * W[h][k][d]   via V_WMMA_F32_16X16X4_F32
//    One wave per (16-node, head) block -> 16x64 output, 4 accumulators.
//    A frag (16x4 f32): lane = {khalf,m}: v0 = A[m][2*khalf], v1 = A[m][2*khalf+1]
//    B frag (4x16 f32): lane = {khalf,n}: v0 = B[2*khalf][n], v1 = B[2*khalf+1][n]
//    C/D (16x16 f32):   vgpr r, lane L:   D[r + 8*(L>=16)][L&15]
// ---------------------------------------------------------------------------
__global__ void gat_gemm_wmma(const float* __restrict__ X,
                              const float* __restrict__ W,
                              float* __restrict__ T) {
    const int lane  = threadIdx.x;        // block = 32 threads (one wave)
    const int tn    = blockIdx.x;         // node tile (16 rows)
    const int h     = blockIdx.y;         // head
    const int m     = lane & 15;
    const int khalf = lane >> 4;          // 0 or 1

    const float* xrow  = X + (size_t)(tn * 16 + m) * IN_F;
    const float* wbase = W + (size_t)h * IN_F * HDIM;

    v8f acc0 = {}, acc1 = {}, acc2 = {}, acc3 = {};
    for (int kk = 0; kk < IN_F; kk += 4) {
        const int ka = kk + khalf * 2;
        v2f a; a.x = xrow[ka]; a.y = xrow[ka + 1];
        const float* w0 = wbase + (size_t)ka * HDIM + m;
        const float* w1 = w0 + HDIM;
        v2f b0; b0.x = w0[0];  b0.y = w1[0];
        v2f b1; b1.x = w0[16]; b1.y = w1[16];
        v2f b2; b2.x = w0[32]; b2.y = w1[32];
        v2f b3; b3.x = w0[48]; b3.y = w1[48];
        acc0 = __builtin_amdgcn_wmma_f32_16x16x4_f32(false, a, false, b0, (short)0, acc0, false, false);
        acc1 = __builtin_amdgcn_wmma_f32_16x16x4_f32(false, a, false, b1, (short)0, acc1, false, false);
        acc2 = __builtin_amdgcn_wmma_f32_16x16x4_f32(false, a, false, b2, (short)0, acc2, false, false);
        acc3 = __builtin_amdgcn_wmma_f32_16x16x4_f32(false, a, false, b3, (short)0, acc3, false, false);
    }

    const int nc    = lane & 15;
    const int rbase = tn * 16 + (lane >> 4) * 8;
    float* tb = T + ((size_t)h * NNODES + rbase) * HDIM + nc;
#pragma unroll
    for (int r = 0; r < 8; r++) {
        tb[(size_t)r * HDIM + 0]  = acc0[r];
        tb[(size_t)r * HDIM + 16] = acc1[r];
        tb[(size_t)r * HDIM + 32] = acc2[r];
        tb[(size_t)r * HDIM + 48] = acc3[r];
    }
}

// ---------------------------------------------------------------------------
// 2) per-node attention pre-dots: asrc[h][n] = t[h,n,:].a[h,:64]; adst likewise
// ---------------------------------------------------------------------------
__global__ void gat_alpha(const float* __restrict__ T,
                          const float* __restrict__ av,
                          float* __restrict__ asrc,
                          float* __restrict__ adst) {
    const int i = blockIdx.x * 256 + threadIdx.x;
    if (i >= NH * NNODES) return;
    const int h = i / NNODES;
    const int n = i % NNODES;
    const float* trow = T + ((size_t)h * NNODES + n) * HDIM;
    const float* ah   = av + h * 2 * HDIM;
    float s0 = 0.f, s1 = 0.f;
#pragma unroll 8
    for (int d = 0; d < HDIM; d++) {
        const float t = trow[d];
        s0 += t * ah[d];
        s1 += t * ah[HDIM + d];
    }
    asrc[i] = s0;
    adst[i] = s1;
}

// ---------------------------------------------------------------------------
// 3) per-edge scores + leaky relu + per-head global max (ordered-uint atomics)
// ---------------------------------------------------------------------------
__global__ void gat_scores(const long long* __restrict__ ei,
                           const float* __restrict__ asrc,
                           const float* __restrict__ adst,
                           float* __restrict__ scores,
                           unsigned* __restrict__ maxb) {
    __shared__ unsigned smax[NH];
    const int tid = threadIdx.x;
    const int e   = blockIdx.x * 256 + tid;
    if (tid < NH) smax[tid] = 0u;
    __syncthreads();
    if (e < NEDGES) {
        const int s = (int)ei[e];
        const int d = (int)ei[NEDGES + e];
#pragma unroll
        for (int h = 0; h < NH; h++) {
            float sc = asrc[h * NNODES + s] + adst[h * NNODES + d];
            sc = (sc >= 0.f) ? sc : NSLOPE * sc;
            scores[(size_t)h * NEDGES + e] = sc;
            atomicMax(&smax[h], f2ord(sc));
        }
    }
    __syncthreads();
    if (tid < NH) atomicMax(&maxb[tid], smax[tid]);
}

// ---------------------------------------------------------------------------
// 4) deterministic per-block partial sums of exp(score - max)
// ---------------------------------------------------------------------------
__global__ void gat_expsum(const float* __restrict__ scores,
                           const unsigned* __restrict__ maxb,
                           float* __restrict__ partial) {
    __shared__ float sdata[256];
    const int tid = threadIdx.x;
    const int e   = blockIdx.x * 256 + tid;
    float vals[NH];
#pragma unroll
    for (int h = 0; h < NH; h++) {
        float v = 0.f;
        if (e < NEDGES) v = expf(scores[(size_t)h * NEDGES + e] - ord2f(maxb[h]));
        vals[h] = v;
    }
#pragma unroll
    for (int h = 0; h < NH; h++) {
        sdata[tid] = vals[h];
        __syncthreads();
        for (int s = 128; s > 0; s >>= 1) {
            if (tid < s) sdata[tid] += sdata[tid + s];
            __syncthreads();
        }
        if (tid == 0) partial[h * gridDim.x + blockIdx.x] = sdata[0];
        __syncthreads();
    }
}

__global__ void gat_sum_reduce(const float* __restrict__ partial,
                               float* __restrict__ sumb, int nblk) {
    const int h = threadIdx.x;
    if (h < NH) {
        float s = 0.f;
        for (int b = 0; b < nblk; b++) s += partial[h * nblk + b];
        sumb[h] = s;
    }
}

// ---------------------------------------------------------------------------
// 5) scatter-add: agg[dst][h*64+d] += w(h,e) * t[h][src][d]
//    one 256-thread block per edge: thread -> (h = tid>>6, d = tid&63)
// ---------------------------------------------------------------------------
__global__ void gat_scatter(const long long* __restrict__ ei,
                            const float* __restrict__ scores,
                            const unsigned* __restrict__ maxb,
                            const float* __restrict__ sumb,
                            const float* __restrict__ T,
                            float* __restrict__ agg) {
    const int e   = blockIdx.x;
    const int tid = threadIdx.x;
    const int h   = tid >> 6;
    const int d   = tid & 63;
    const int s   = (int)ei[e];
    const int dd  = (int)ei[NEDGES + e];
    const float w = expf(scores[(size_t)h * NEDGES + e] - ord2f(maxb[h])) / sumb[h];
    const float val = w * T[((size_t)h * NNODES + s) * HDIM + d];
    atomicAdd(&agg[(size_t)dd * OUT_F + h * HDIM + d], val);
}

// ---------------------------------------------------------------------------
// 6) LayerNorm over 256 features; one wave32 per node, shuffle reduction
// ---------------------------------------------------------------------------
__global__ void gat_layernorm(const float* __restrict__ agg,
                              const float* __restrict__ gamma,
                              const float* __restrict__ beta,
                              float* __restrict__ out) {
    const int wave = threadIdx.x >> 5;
    const int lane = threadIdx.x & 31;
    const int n    = blockIdx.x * 8 + wave;
    if (n >= NNODES) return;
    const float* row = agg + (size_t)n * OUT_F;
    float v[8];
    float s = 0.f, s2 = 0.f;
#pragma unroll
    for (int i = 0; i < 8; i++) {
        v[i] = row[lane + i * 32];
        s  += v[i];
        s2 += v[i] * v[i];
    }
#pragma unroll
    for (int off = 16; off > 0; off >>= 1) {
        s  += __shfl_xor(s,  off, 32);
        s2 += __shfl_xor(s2, off, 32);
    }
    const float mean = s * (1.0f / OUT_F);
    const float var  = s2 * (1.0f / OUT_F) - mean * mean;
    const float inv  = rsqrtf(var + LN_EPSF);
#pragma unroll
    for (int i = 0; i < 8; i++) {
        const int c = lane + i * 32;
        out[(size_t)n * OUT_F + c] = (v[i] - mean) * inv * gamma[c] + beta[c];
    }
}

// ---------------------------------------------------------------------------
extern "C" void kernel_launch(void* const* d_in, const int* in_sizes, int n_in,
                              void* d_out, int out_size, void* d_ws, size_t ws_size,
                              hipStream_t stream) {
    const float*     X     = (const float*)d_in[0];      // [1, N, 256]
    const long long* EI    = (const long long*)d_in[1];  // [2, E] int64
    const float*     W     = (const float*)d_in[2];      // [H, 256, 64]
    const float*     A     = (const float*)d_in[3];      // [H, 128]
    const float*     gamma = (const float*)d_in[4];      // [256]
    const float*     beta  = (const float*)d_in[5];      // [256]
    float*           out   = (float*)d_out;              // [1, N, 256]

    // workspace carve-up (floats): ~44.2 MB total
    float* T       = (float*)d_ws;                        // H*N*HD   = 5,120,000
    float* agg     = T    + (size_t)NH * NNODES * HDIM;   // N*256    = 5,120,000
    float* asrc    = agg  + (size_t)NNODES * OUT_F;       // H*N      =    80,000
    float* adst    = asrc + (size_t)NH * NNODES;          // H*N      =    80,000
    float* scores  = adst + (size_t)NH * NNODES;          // H*E      =   640,000
    const int SBLK = (NEDGES + 255) / 256;                // 625
    float* partial = scores + (size_t)NH * NEDGES;        // H*SBLK   =     2,500
    unsigned* maxb = (unsigned*)(partial + NH * SBLK);    // NH
    float* sumb    = (float*)(maxb + NH);                 // NH

    gat_init<<<(NNODES * OUT_F + 255) / 256, 256, 0, stream>>>(agg, maxb);
    gat_gemm_wmma<<<dim3(NNODES / 16, NH), 32, 0, stream>>>(X, W, T);
    gat_alpha<<<(NH * NNODES + 255) / 256, 256, 0, stream>>>(T, A, asrc, adst);
    gat_scores<<<SBLK, 256, 0, stream>>>(EI, asrc, adst, scores, maxb);
    gat_expsum<<<SBLK, 256, 0, stream>>>(scores, maxb, partial);
    gat_sum_reduce<<<1, 32, 0, stream>>>(partial, sumb, SBLK);
    gat_scatter<<<NEDGES, 256, 0, stream>>>(EI, scores, maxb, sumb, T, agg);
    gat_layernorm<<<(NNODES + 7) / 8, 256, 0, stream>>>(agg, gamma, beta, out);
}